// CCPL_85461259256050
// MI455X (gfx1250) — compile-verified
//
#include <hip/hip_runtime.h>
#include <math.h>

// CDNA5 / gfx1250 fused PatchNCE loss.
//
// - All GEMMs run on V_WMMA_F32_16X16X4_F32 (the only fp32 WMMA; full fp32
//   precision matching the reference).
// - Large GEMMs stage weight tiles in LDS via the gfx1250 async-to-LDS path
//   (global_load_async_to_lds_b128 + s_wait_asynccnt), double-buffered so the
//   async DMA of the next K-chunk overlaps WMMA on the current one, and so a
//   64x64 weight chunk is fetched once per 8-wave block instead of per wave.
// - The 512x512 NCE logits matrices are never materialized: log-sum-exp is
//   computed online in the WMMA accumulators + a 16-lane shfl butterfly.
//
// Workspace (~20.1 MB):
//   X    : 4096*512 f32 | Hbuf : 4096*512 f32 | Fq,Fk : 4096*128 f32 each
//   rowLoss : 4*4096 f32

typedef __attribute__((ext_vector_type(2))) float v2f;
typedef __attribute__((ext_vector_type(8))) float v8f;

#define NCE_TAU 0.07f

// ---- gfx1250 async-to-LDS helpers (inline asm; no portable builtin) -------
__device__ __forceinline__ void async_load_b128(unsigned int ldsByteAddr,
                                                const float* gaddr) {
    asm volatile("global_load_async_to_lds_b128 %0, %1, off"
                 :: "v"(ldsByteAddr), "v"(gaddr)
                 : "memory");
}
__device__ __forceinline__ void wait_async0() {
    asm volatile("s_wait_asynccnt 0x0" ::: "memory");
}

// ---------------------------------------------------------------------------
// 1) Gather: X[row][c] = feat[b,c,h0+dh,w0+dw] - feat[b,c,h0+1,w0+1]
// ---------------------------------------------------------------------------
__global__ void gather_diff_kernel(const float* __restrict__ feat,
                                   const int* __restrict__ sid,
                                   float* __restrict__ X,
                                   int C, int HW) {
    int idx = blockIdx.x * blockDim.x + threadIdx.x;
    int total = 4096 * C;
    if (idx >= total) return;
    int c   = idx % C;
    int row = idx / C;
    int b = row >> 9;
    int t = row & 511;
    int s = t >> 3;
    int n = t & 7;
    int h0 = sid[2 * s + 0];
    int w0 = sid[2 * s + 1];
    int dh = (0x22211000u >> (4 * n)) & 0xF;   // DH = {0,0,0,1,1,2,2,2}
    int dw = (0x21020210u >> (4 * n)) & 0xF;   // DW = {0,1,2,0,2,0,1,2}
    const float* fb = feat + (size_t)(b * C + c) * HW * HW;
    float nv = fb[(h0 + dh) * HW + (w0 + dw)];
    float cv = fb[(h0 + 1) * HW + (w0 + 1)];
    X[(size_t)row * C + c] = nv - cv;
}

// ---------------------------------------------------------------------------
// 2a) LDS-staged WMMA GEMM: Out(4096 x N) = A(4096 x K) * Wt(N x K)^T + bias.
//     Block = 256 threads (8 waves) -> 128 rows x 64 cols output tile.
//     Weight chunks (64 cols x 64 k, 16KB, row stride 68 floats for
//     bank-conflict-free ds_load_b64) staged by async-to-LDS, double-buffered.
//     Requires N % 64 == 0, K % 64 == 0.
// ---------------------------------------------------------------------------
#define BSTRIDE 68

__global__ void wmma_gemm_lds_kernel(const float* __restrict__ A,
                                     const float* __restrict__ Wt,
                                     const float* __restrict__ bias,
                                     float* __restrict__ Out,
                                     int N, int K, int doRelu) {
    __shared__ float bt[2][64 * BSTRIDE];

    const int tid  = threadIdx.x;          // 0..255
    const int lane = tid & 31;
    const int wave = tid >> 5;             // 0..7
    const int half = lane >> 4;
    const int m16  = lane & 15;
    const int rowBase  = blockIdx.y * 128 + wave * 16;
    const int colBase0 = blockIdx.x * 64;
    const int nk = K >> 6;                 // K / 64 chunks

    // LDS byte base of each buffer (low 32 bits of a shared pointer are the
    // wave-relative LDS address on gfx1250 -- LDS aperture truncation rule).
    unsigned int ldsBase0 = (unsigned int)(unsigned long long)(&bt[0][0]);
    unsigned int ldsBase1 = (unsigned int)(unsigned long long)(&bt[1][0]);

    // Stage chunk kc of the 64 x 64 weight tile into buffer buf.
    // 64 rows x 16 b128-segments = 1024 segments, 4 per thread.
    auto stage = [&](int buf, int kc) {
        unsigned int base = buf ? ldsBase1 : ldsBase0;
        const float* wsrc = Wt + (size_t)colBase0 * K + kc * 64;
#pragma unroll
        for (int t = 0; t < 4; ++t) {
            int si   = t * 256 + tid;
            int r    = si >> 4;            // 0..63 (weight row = output col)
            int cseg = (si & 15) << 2;     // 0,4,...,60
            const float* g = wsrc + (size_t)r * K + cseg;
            unsigned int l = base + (unsigned int)(r * BSTRIDE + cseg) * 4u;
            async_load_b128(l, g);
        }
    };

    v8f acc[4];
#pragma unroll
    for (int i = 0; i < 4; ++i) acc[i] = (v8f){0.f,0.f,0.f,0.f,0.f,0.f,0.f,0.f};

    const float* Arow = A + (size_t)(rowBase + m16) * K + 2 * half;

    stage(0, 0);
    for (int kc = 0; kc < nk; ++kc) {
        wait_async0();
        __syncthreads();                        // chunk kc visible to all waves
        if (kc + 1 < nk) stage((kc + 1) & 1, kc + 1);

        // Prefetch this wave's A fragments for the whole chunk (16 k-steps).
        v2f af[16];
        const float* Ac = Arow + kc * 64;
#pragma unroll
        for (int kk = 0; kk < 16; ++kk) af[kk] = *(const v2f*)(Ac + 4 * kk);

        const float* bbuf = &bt[kc & 1][0];
#pragma unroll
        for (int kk = 0; kk < 16; ++kk) {
#pragma unroll
            for (int i = 0; i < 4; ++i) {
                v2f bf = *(const v2f*)(bbuf + (i * 16 + m16) * BSTRIDE +
                                       4 * kk + 2 * half);
                acc[i] = __builtin_amdgcn_wmma_f32_16x16x4_f32(
                    false, af[kk], false, bf, (short)0, acc[i], false, false);
            }
        }
        __syncthreads();                        // all reads done before reuse
    }

#pragma unroll
    for (int i = 0; i < 4; ++i) {
        int colBase = colBase0 + i * 16;
        float bvi = bias[colBase + m16];
#pragma unroll
        for (int v = 0; v < 8; ++v) {
            float x = acc[i][v] + bvi;
            if (doRelu) x = fmaxf(x, 0.0f);
            Out[(size_t)(rowBase + v + 8 * half) * N + colBase + m16] = x;
        }
    }
}

// ---------------------------------------------------------------------------
// 2b) Simple per-wave WMMA GEMM for small N (16/32). One wave per
//     16 x (16*CT) tile. f32 fragment layout per ISA 7.12.2.
// ---------------------------------------------------------------------------
template <int CT>
__global__ void wmma_gemm_kernel(const float* __restrict__ A,
                                 const float* __restrict__ Wt,
                                 const float* __restrict__ bias,
                                 float* __restrict__ Out,
                                 int N, int K, int doRelu) {
    const int lane = threadIdx.x;
    const int half = lane >> 4;
    const int m16  = lane & 15;
    const int rowBase  = blockIdx.y * 16;
    const int colBase0 = blockIdx.x * 16 * CT;

    v8f acc[CT];
#pragma unroll
    for (int i = 0; i < CT; ++i) acc[i] = (v8f){0.f,0.f,0.f,0.f,0.f,0.f,0.f,0.f};

    const float* Arow = A + (size_t)(rowBase + m16) * K + 2 * half;
    for (int k = 0; k < K; k += 4) {
        v2f a = *(const v2f*)(Arow + k);
#pragma unroll
        for (int i = 0; i < CT; ++i) {
            const float* Brow = Wt + (size_t)(colBase0 + i * 16 + m16) * K + 2 * half;
            v2f bf = *(const v2f*)(Brow + k);
            acc[i] = __builtin_amdgcn_wmma_f32_16x16x4_f32(
                false, a, false, bf, (short)0, acc[i], false, false);
        }
    }

#pragma unroll
    for (int i = 0; i < CT; ++i) {
        int colBase = colBase0 + i * 16;
        float bvi = bias[colBase + m16];
#pragma unroll
        for (int v = 0; v < 8; ++v) {
            float x = acc[i][v] + bvi;
            if (doRelu) x = fmaxf(x, 0.0f);
            Out[(size_t)(rowBase + v + 8 * half) * N + colBase + m16] = x;
        }
    }
}

// ---------------------------------------------------------------------------
// 3) Row L2-normalize in place.
// ---------------------------------------------------------------------------
__global__ void row_normalize_kernel(float* __restrict__ F, int Cp) {
    __shared__ float red[128];
    int row = blockIdx.x;
    int tid = threadIdx.x;  // 128 threads, Cp <= 128
    float val = 0.0f, sq = 0.0f;
    if (tid < Cp) { val = F[(size_t)row * Cp + tid]; sq = val * val; }
    red[tid] = sq;
    __syncthreads();
    for (int s = 64; s > 0; s >>= 1) {
        if (tid < s) red[tid] += red[tid + s];
        __syncthreads();
    }
    float scale = 1.0f / (sqrtf(red[0]) + 1e-7f);
    if (tid < Cp) F[(size_t)row * Cp + tid] = val * scale;
}

// ---------------------------------------------------------------------------
// 4) NCE row loss: WMMA Gram tiles + online LSE in accumulators.
// ---------------------------------------------------------------------------
template <int CP>
__global__ void nce_rowloss_kernel(const float* __restrict__ Fq,
                                   const float* __restrict__ Fk,
                                   float* __restrict__ rowLoss) {
    __shared__ float lseS[16];
    const int lane = threadIdx.x;
    const int half = lane >> 4;
    const int m16  = lane & 15;
    const int b    = blockIdx.y;
    const int rowT = blockIdx.x * 16;
    const float invTau = 1.0f / NCE_TAU;

    v2f af[CP / 4];
    const float* aRow = Fq + (size_t)(b * 512 + rowT + m16) * CP + 2 * half;
#pragma unroll
    for (int kk = 0; kk < CP / 4; ++kk) af[kk] = *(const v2f*)(aRow + 4 * kk);

    float mx[8], sm[8];
#pragma unroll
    for (int v = 0; v < 8; ++v) { mx[v] = -1e30f; sm[v] = 0.0f; }

    for (int ct = 0; ct < 32; ++ct) {
        const float* bRow = Fk + (size_t)(b * 512 + ct * 16 + m16) * CP + 2 * half;
        v8f acc = (v8f){0.f,0.f,0.f,0.f,0.f,0.f,0.f,0.f};
#pragma unroll
        for (int kk = 0; kk < CP / 4; ++kk) {
            v2f bf = *(const v2f*)(bRow + 4 * kk);
            acc = __builtin_amdgcn_wmma_f32_16x16x4_f32(
                false, af[kk], false, bf, (short)0, acc, false, false);
        }
#pragma unroll
        for (int v = 0; v < 8; ++v) {
            float val = acc[v] * invTau;
            float nm  = fmaxf(mx[v], val);
            sm[v] = sm[v] * __expf(mx[v] - nm) + __expf(val - nm);
            mx[v] = nm;
        }
    }

#pragma unroll
    for (int v = 0; v < 8; ++v) {
#pragma unroll
        for (int mask = 1; mask < 16; mask <<= 1) {
            float om = __shfl_xor(mx[v], mask);
            float os = __shfl_xor(sm[v], mask);
            float nm = fmaxf(mx[v], om);
            sm[v] = sm[v] * __expf(mx[v] - nm) + os * __expf(om - nm);
            mx[v] = nm;
        }
        if (m16 == v) lseS[8 * half + v] = mx[v] + __logf(sm[v]);
    }
    __syncthreads();

    if (lane < 16) {
        int grow = b * 512 + rowT + lane;
        const float* q = Fq + (size_t)grow * CP;
        const float* k = Fk + (size_t)grow * CP;
        float d = 0.0f;
#pragma unroll
        for (int c = 0; c < CP; ++c) d += q[c] * k[c];
        rowLoss[grow] = lseS[lane] - d * invTau;
    }
}

// ---------------------------------------------------------------------------
// 5) Deterministic final reduction: loss = sum(rowLoss) / 4096
// ---------------------------------------------------------------------------
__global__ void final_reduce_kernel(const float* __restrict__ rowLoss,
                                    float* __restrict__ out) {
    __shared__ float red[256];
    int tid = threadIdx.x;
    float s = 0.0f;
    for (int i = tid; i < 4 * 4096; i += 256) s += rowLoss[i];
    red[tid] = s;
    __syncthreads();
    for (int st = 128; st > 0; st >>= 1) {
        if (tid < st) red[tid] += red[tid + st];
        __syncthreads();
    }
    if (tid == 0) out[0] = red[0] * (1.0f / 4096.0f);
}

extern "C" void kernel_launch(void* const* d_in, const int* in_sizes, int n_in,
                              void* d_out, int out_size, void* d_ws, size_t ws_size,
                              hipStream_t stream) {
    (void)in_sizes; (void)n_in; (void)out_size; (void)ws_size;
    static const int Cs[4]  = {64, 128, 256, 512};
    static const int HWs[4] = {256, 128, 64, 32};

    float* X       = (float*)d_ws;            // 4096*512
    float* Hb      = X  + 4096 * 512;         // 4096*512
    float* Fq      = Hb + 4096 * 512;         // 4096*128
    float* Fk      = Fq + 4096 * 128;         // 4096*128
    float* rowLoss = Fk + 4096 * 128;         // 4*4096
    float* out     = (float*)d_out;

    for (int l = 0; l < 4; ++l) {
        int C = Cs[l], HW = HWs[l], Cp = C / 4;
        // setup_inputs dict order: per layer {fq, fk, sid, w1, b1, w2, b2}
        const float* fq  = (const float*)d_in[7 * l + 0];
        const float* fk  = (const float*)d_in[7 * l + 1];
        const int*   sid = (const int*)  d_in[7 * l + 2];
        const float* w1  = (const float*)d_in[7 * l + 3];
        const float* b1  = (const float*)d_in[7 * l + 4];
        const float* w2  = (const float*)d_in[7 * l + 5];
        const float* b2  = (const float*)d_in[7 * l + 6];

        const float* feats[2] = {fq, fk};
        float*       Fs[2]    = {Fq, Fk};
        for (int t = 0; t < 2; ++t) {
            int total = 4096 * C;
            gather_diff_kernel<<<(total + 255) / 256, 256, 0, stream>>>(
                feats[t], sid, X, C, HW);
            // GEMM1: N = K = C, always multiples of 64 -> LDS-staged kernel.
            wmma_gemm_lds_kernel<<<dim3(C / 64, 32), 256, 0, stream>>>(
                X, w1, b1, Hb, C, C, 1);
            // GEMM2: N = C/4 in {16,32,64,128}
            if (Cp % 64 == 0)
                wmma_gemm_lds_kernel<<<dim3(Cp / 64, 32), 256, 0, stream>>>(
                    Hb, w2, b2, Fs[t], Cp, C, 0);
            else if (Cp % 32 == 0)
                wmma_gemm_kernel<2><<<dim3(Cp / 32, 256), 32, 0, stream>>>(
                    Hb, w2, b2, Fs[t], Cp, C, 0);
            else
                wmma_gemm_kernel<1><<<dim3(Cp / 16, 256), 32, 0, stream>>>(
                    Hb, w2, b2, Fs[t], Cp, C, 0);
            row_normalize_kernel<<<4096, 128, 0, stream>>>(Fs[t], Cp);
        }
        switch (Cp) {
            case 16:
                nce_rowloss_kernel<16><<<dim3(32, 8), 32, 0, stream>>>(Fq, Fk, rowLoss + l * 4096);
                break;
            case 32:
                nce_rowloss_kernel<32><<<dim3(32, 8), 32, 0, stream>>>(Fq, Fk, rowLoss + l * 4096);
                break;
            case 64:
                nce_rowloss_kernel<64><<<dim3(32, 8), 32, 0, stream>>>(Fq, Fk, rowLoss + l * 4096);
                break;
            default:
                nce_rowloss_kernel<128><<<dim3(32, 8), 32, 0, stream>>>(Fq, Fk, rowLoss + l * 4096);
                break;
        }
    }
    final_reduce_kernel<<<1, 256, 0, stream>>>(rowLoss, out);
}